// MPNNPORE_44367012168465
// MI455X (gfx1250) — compile-verified
//
#include <hip/hip_runtime.h>
#include <cstdint>

// ---------------------------------------------------------------------------
// MPNN-pore forward pass for gfx1250 (MI455X), f32 WMMA 16x16x4 throughout.
// B=4, N=128, Np=16, E=1280, Esp=1024, D=O=32, Fin=80, C=16, M=64, T=3.
// ---------------------------------------------------------------------------

typedef __attribute__((ext_vector_type(2))) float v2f;
typedef __attribute__((ext_vector_type(8))) float v8f;

#define LSLOPE 0.01f

__device__ __forceinline__ float lrelu(float x) { return x > 0.0f ? x : LSLOPE * x; }
__device__ __forceinline__ float sigm(float x) { return 1.0f / (1.0f + __expf(-x)); }
__device__ __forceinline__ v8f zero8() {
  v8f z = {0.f, 0.f, 0.f, 0.f, 0.f, 0.f, 0.f, 0.f};
  return z;
}
// D(16x16,f32) = A(16x4,f32) x B(4x16,f32) + C   (wave32, 20 k-steps for K=80)
__device__ __forceinline__ v8f wmma4(v2f a, v2f b, v8f c) {
  return __builtin_amdgcn_wmma_f32_16x16x4_f32(false, a, false, b, (short)0, c,
                                               false, false);
}

// ---------------------------------------------------------------------------
// Kernel 1: site / pore embeddings (elementwise).
// s[b,n,d]  = lrelu(sites[b,n,0]*seW[d] + seB[d])
// sp[b,n,d] = lrelu(sp0*sepW[0,d] + sp1*sepW[1,d] + sepB[d])
// ---------------------------------------------------------------------------
__global__ __launch_bounds__(256) void MPNN_embed_sites(
    const float* __restrict__ sites, const float* __restrict__ sites_p,
    const float* __restrict__ seW, const float* __restrict__ seB,
    const float* __restrict__ sepW, const float* __restrict__ sepB,
    float* __restrict__ s, float* __restrict__ sp) {
  int i = blockIdx.x * 256 + threadIdx.x;
  const int NS = 4 * 128 * 32;
  const int NSP = 4 * 16 * 32;
  if (i < NS) {
    int d = i & 31, bn = i >> 5;
    s[i] = lrelu(sites[bn] * seW[d] + seB[d]);
  } else if (i < NS + NSP) {
    int j = i - NS;
    int d = j & 31, bn = j >> 5;
    sp[j] = lrelu(sites_p[bn * 2] * sepW[d] + sites_p[bn * 2 + 1] * sepW[32 + d] +
                  sepB[d]);
  }
}

// ---------------------------------------------------------------------------
// Kernel 2: edge embeddings. Gaussian RBF (16 centers) computed in-register as
// the WMMA A fragment; x W[16x16] + bias, LeakyReLU.  One wave per 16 edges.
// Segments: [0,320) be, [320,576) bsp, [576,832) bps  (flat over B*E edges).
// ---------------------------------------------------------------------------
__global__ __launch_bounds__(256) void MPNN_embed_edges(
    const float* __restrict__ bonds, const float* __restrict__ bonds_sp,
    const float* __restrict__ bonds_ps, const float* __restrict__ eeW,
    const float* __restrict__ eeB, const float* __restrict__ eepW,
    const float* __restrict__ eepB, float* __restrict__ be,
    float* __restrict__ bsp, float* __restrict__ bps) {
  int wave = threadIdx.x >> 5, lane = threadIdx.x & 31;
  int g = blockIdx.x * 8 + wave;
  const float* dsrc;
  const float* W;
  const float* bias;
  float* out;
  int tloc;
  if (g < 320) {
    dsrc = bonds; W = eeW; bias = eeB; out = be; tloc = g;
  } else if (g < 576) {
    dsrc = bonds_sp; W = eepW; bias = eepB; out = bsp; tloc = g - 320;
  } else {
    dsrc = bonds_ps; W = eepW; bias = eepB; out = bps; tloc = g - 576;
  }
  int e0 = tloc * 16;
  int row = lane & 15, half = lane >> 4;
  float d = dsrc[e0 + row];
  const float MUSTEP = 10.0f / 15.0f;  // linspace(0,10,16) step
  v8f acc = zero8();
  for (int k0 = 0; k0 < 16; k0 += 4) {
    int ka = k0 + 2 * half;
    float t0 = d - (float)ka * MUSTEP;
    float t1 = d - (float)(ka + 1) * MUSTEP;
    v2f a, bb;
    a.x = __expf(-t0 * t0);
    a.y = __expf(-t1 * t1);
    bb.x = W[ka * 16 + row];
    bb.y = W[(ka + 1) * 16 + row];
    acc = wmma4(a, bb, acc);
  }
  float bv = bias[row];
#pragma unroll
  for (int r = 0; r < 8; ++r)
    out[(size_t)(e0 + r + 8 * half) * 16 + row] = lrelu(acc[r] + bv);
}

// ---------------------------------------------------------------------------
// Kernel 3: equivariant edge message + attention gate + scatter-add.
// lat = lrelu(v@W1 + v@W2 + b); gate = sigmoid(lat.aw + ab); m[recv] += g*lat.
// One block per batch; src/dst features, W1, W2, m accumulator all in LDS.
// v rows are gathered straight from LDS while building A fragments.
// ---------------------------------------------------------------------------
__global__ __launch_bounds__(256) void MPNN_msg(
    const float* __restrict__ srcF, const float* __restrict__ dstF,
    const float* __restrict__ edgeF, const int* __restrict__ idx1g,
    const int* __restrict__ idx2g, const float* __restrict__ W1g,
    const float* __restrict__ W2g, const float* __restrict__ biasg,
    const float* __restrict__ awg, const float* __restrict__ abg,
    float* __restrict__ mOut, int E, int Nsrc, int Ndst, int Nrecv) {
  __shared__ float sd[144 * 32];     // src rows, then dst rows (if distinct)
  __shared__ float m_lds[128 * 32];  // receiver accumulator
  __shared__ float w1[80 * 32];
  __shared__ float w2[80 * 32];
  __shared__ float bias_s[32];
  __shared__ float aw_s[32];

  int b = blockIdx.x;
  int tid = threadIdx.x;
  bool same = (srcF == dstF);
  int dstBase = same ? 0 : Nsrc * 32;

  for (int i = tid; i < Nsrc * 32; i += 256) sd[i] = srcF[(size_t)b * Nsrc * 32 + i];
  if (!same)
    for (int i = tid; i < Ndst * 32; i += 256)
      sd[dstBase + i] = dstF[(size_t)b * Ndst * 32 + i];
  for (int i = tid; i < Nrecv * 32; i += 256) m_lds[i] = 0.0f;
  for (int i = tid; i < 80 * 32; i += 256) {
    w1[i] = W1g[i];
    w2[i] = W2g[i];
  }
  if (tid < 32) {
    bias_s[tid] = biasg[tid];
    aw_s[tid] = awg[tid];
  }
  __syncthreads();

  float abv = abg[0];
  int wave = tid >> 5, lane = tid & 31;
  int row = lane & 15, half = lane >> 4, col = row;
  int tiles = E >> 4;

  for (int tile = wave; tile < tiles; tile += 8) {
    int e0 = tile << 4;
    int i1 = idx1g[e0 + row];
    int i2 = idx2g[e0 + row];
    const float* ef = edgeF + ((size_t)b * E + e0 + row) * 16;

    v8f aL0 = zero8(), aL1 = zero8(), aS0 = zero8(), aS1 = zero8();
    for (int k0 = 0; k0 < 80; k0 += 4) {  // Fin = 32(src) + 32(dst) + 16(edge)
      int ka = k0 + 2 * half;             // region is uniform within a k-step
      float x0, x1;
      if (ka < 32) {
        x0 = sd[i1 * 32 + ka];
        x1 = sd[i1 * 32 + ka + 1];
      } else if (ka < 64) {
        x0 = sd[dstBase + i2 * 32 + ka - 32];
        x1 = sd[dstBase + i2 * 32 + ka - 31];
      } else {
        x0 = ef[ka - 64];
        x1 = ef[ka - 63];
      }
      v2f a;
      a.x = x0;
      a.y = x1;
      v2f b10, b11, b20, b21;
      b10.x = w1[ka * 32 + col];       b10.y = w1[(ka + 1) * 32 + col];
      b11.x = w1[ka * 32 + 16 + col];  b11.y = w1[(ka + 1) * 32 + 16 + col];
      b20.x = w2[ka * 32 + col];       b20.y = w2[(ka + 1) * 32 + col];
      b21.x = w2[ka * 32 + 16 + col];  b21.y = w2[(ka + 1) * 32 + 16 + col];
      aL0 = wmma4(a, b10, aL0);  // y_loc, outs 0..15
      aL1 = wmma4(a, b11, aL1);  // y_loc, outs 16..31
      aS0 = wmma4(a, b20, aS0);  // y_sum, outs 0..15
      aS1 = wmma4(a, b21, aS1);  // y_sum, outs 16..31
    }

    // lat + attention gate (butterfly reduce within each 16-lane half-group)
    float g[8];
#pragma unroll
    for (int r = 0; r < 8; ++r) {
      float l0 = lrelu(aL0[r] + aS0[r] + bias_s[col]);
      float l1 = lrelu(aL1[r] + aS1[r] + bias_s[16 + col]);
      aL0[r] = l0;
      aL1[r] = l1;
      float p = l0 * aw_s[col] + l1 * aw_s[16 + col];
      p += __shfl_xor(p, 1, 32);
      p += __shfl_xor(p, 2, 32);
      p += __shfl_xor(p, 4, 32);
      p += __shfl_xor(p, 8, 32);
      g[r] = sigm(p + abv);
    }
    // scatter-add gated messages to receivers (LDS f32 atomics)
#pragma unroll
    for (int r = 0; r < 8; ++r) {
      int ridx = idx2g[e0 + r + 8 * half];
      atomicAdd(&m_lds[ridx * 32 + col], g[r] * aL0[r]);
      atomicAdd(&m_lds[ridx * 32 + 16 + col], g[r] * aL1[r]);
    }
  }
  __syncthreads();
  for (int i = tid; i < Nrecv * 32; i += 256)
    mOut[(size_t)b * Nrecv * 32 + i] = m_lds[i];
}

// ---------------------------------------------------------------------------
// Kernel 4: residual node updates (sites blocks 0..3, pores blocks 4..7).
// sites: h=[s|m|m_ps] (128x96) -> lrelu(h@W1+b1) -> lrelu(@W2+b2) -> s += .
// pores: hp=[sp|m_sp] (16x64)  -> same with nup weights -> sp += .
// 64KB LDS carve: h(48KB)+h1(16KB); weights streamed from global (L2-hot).
// ---------------------------------------------------------------------------
__global__ __launch_bounds__(256) void MPNN_update(
    float* __restrict__ s, float* __restrict__ sp, const float* __restrict__ m,
    const float* __restrict__ mps, const float* __restrict__ msp,
    const float* __restrict__ nuW1, const float* __restrict__ nub1,
    const float* __restrict__ nuW2, const float* __restrict__ nub2,
    const float* __restrict__ npW1, const float* __restrict__ npb1,
    const float* __restrict__ npW2, const float* __restrict__ npb2) {
  __shared__ float smem[16384];  // 64KB, carved per path
  int tid = threadIdx.x;
  int wave = tid >> 5, lane = tid & 31;
  int row = lane & 15, half = lane >> 4, col = row;

  if (blockIdx.x < 4) {  // ---- site update, batch b ----
    int b = blockIdx.x;
    float* h = smem;           // 128*96
    float* h1 = smem + 12288;  // 128*32
    size_t base = (size_t)b * 128 * 32;
    for (int i = tid; i < 128 * 96; i += 256) {
      int n = i / 96, k = i - n * 96;
      float v;
      if (k < 32) v = s[base + n * 32 + k];
      else if (k < 64) v = m[base + n * 32 + (k - 32)];
      else v = mps[base + n * 32 + (k - 64)];
      h[i] = v;
    }
    __syncthreads();
    for (int j = wave; j < 16; j += 8) {  // 8 m-tiles x 2 n-tiles
      int mt = j >> 1, nt = j & 1;
      v8f acc = zero8();
      for (int k0 = 0; k0 < 96; k0 += 4) {
        int ka = k0 + 2 * half;
        v2f a, bb;
        a.x = h[(mt * 16 + row) * 96 + ka];
        a.y = h[(mt * 16 + row) * 96 + ka + 1];
        bb.x = nuW1[ka * 32 + nt * 16 + col];
        bb.y = nuW1[(ka + 1) * 32 + nt * 16 + col];
        acc = wmma4(a, bb, acc);
      }
      float bv = nub1[nt * 16 + col];
#pragma unroll
      for (int r = 0; r < 8; ++r)
        h1[(mt * 16 + r + 8 * half) * 32 + nt * 16 + col] = lrelu(acc[r] + bv);
    }
    __syncthreads();
    for (int j = wave; j < 16; j += 8) {
      int mt = j >> 1, nt = j & 1;
      v8f acc = zero8();
      for (int k0 = 0; k0 < 32; k0 += 4) {
        int ka = k0 + 2 * half;
        v2f a, bb;
        a.x = h1[(mt * 16 + row) * 32 + ka];
        a.y = h1[(mt * 16 + row) * 32 + ka + 1];
        bb.x = nuW2[ka * 32 + nt * 16 + col];
        bb.y = nuW2[(ka + 1) * 32 + nt * 16 + col];
        acc = wmma4(a, bb, acc);
      }
      float bv = nub2[nt * 16 + col];
#pragma unroll
      for (int r = 0; r < 8; ++r) {
        int nr = mt * 16 + r + 8 * half;
        s[base + nr * 32 + nt * 16 + col] =
            h[nr * 96 + nt * 16 + col] + lrelu(acc[r] + bv);
      }
    }
  } else {  // ---- pore update, batch b ----
    int b = blockIdx.x - 4;
    float* hp = smem;          // 16*64
    float* h1p = smem + 1024;  // 16*32
    size_t base = (size_t)b * 16 * 32;
    for (int i = tid; i < 16 * 64; i += 256) {
      int n = i >> 6, k = i & 63;
      hp[i] = (k < 32) ? sp[base + n * 32 + k] : msp[base + n * 32 + (k - 32)];
    }
    __syncthreads();
    if (wave < 2) {
      int nt = wave;
      v8f acc = zero8();
      for (int k0 = 0; k0 < 64; k0 += 4) {
        int ka = k0 + 2 * half;
        v2f a, bb;
        a.x = hp[row * 64 + ka];
        a.y = hp[row * 64 + ka + 1];
        bb.x = npW1[ka * 32 + nt * 16 + col];
        bb.y = npW1[(ka + 1) * 32 + nt * 16 + col];
        acc = wmma4(a, bb, acc);
      }
      float bv = npb1[nt * 16 + col];
#pragma unroll
      for (int r = 0; r < 8; ++r)
        h1p[(r + 8 * half) * 32 + nt * 16 + col] = lrelu(acc[r] + bv);
    }
    __syncthreads();
    if (wave < 2) {
      int nt = wave;
      v8f acc = zero8();
      for (int k0 = 0; k0 < 32; k0 += 4) {
        int ka = k0 + 2 * half;
        v2f a, bb;
        a.x = h1p[row * 32 + ka];
        a.y = h1p[row * 32 + ka + 1];
        bb.x = npW2[ka * 32 + nt * 16 + col];
        bb.y = npW2[(ka + 1) * 32 + nt * 16 + col];
        acc = wmma4(a, bb, acc);
      }
      float bv = npb2[nt * 16 + col];
#pragma unroll
      for (int r = 0; r < 8; ++r) {
        int nr = r + 8 * half;
        sp[base + nr * 32 + nt * 16 + col] =
            hp[nr * 64 + nt * 16 + col] + lrelu(acc[r] + bv);
      }
    }
  }
}

// ---------------------------------------------------------------------------
// Kernel 5: prediction head (tiny; scalar). thread = (b, o) pair, 4x64 = 256.
// ---------------------------------------------------------------------------
__global__ __launch_bounds__(256) void MPNN_head(
    const float* __restrict__ s, const float* __restrict__ p1W,
    const float* __restrict__ p1b, const float* __restrict__ p2aW,
    const float* __restrict__ p2ab, const float* __restrict__ p2bW,
    const float* __restrict__ p2bb, const float* __restrict__ p3W,
    const float* __restrict__ p3b, float* __restrict__ out) {
  __shared__ float x1[4 * 64];
  __shared__ float x2[4 * 64];
  int tid = threadIdx.x;
  int b = tid >> 6, o = tid & 63;

  float acc = 0.f;
  for (int n = 0; n < 128; ++n) {
    const float* srow = s + ((size_t)b * 128 + n) * 32;
    float t = p1b[o];
    for (int d = 0; d < 32; ++d) t += srow[d] * p1W[d * 64 + o];
    acc += lrelu(t);  // lrelu before the sum-pool
  }
  x1[b * 64 + o] = acc;
  __syncthreads();
  float t = p2ab[o];
  for (int k = 0; k < 64; ++k) t += x1[b * 64 + k] * p2aW[k * 64 + o];
  x2[b * 64 + o] = lrelu(t);
  __syncthreads();
  t = p2bb[o];
  for (int k = 0; k < 64; ++k) t += x2[b * 64 + k] * p2bW[k * 64 + o];
  x1[b * 64 + o] = lrelu(t);
  __syncthreads();
  if (o == 0) {
    float r = p3b[0];
    for (int k = 0; k < 64; ++k) r += x1[b * 64 + k] * p3W[k];
    out[b] = r;
  }
}

// ---------------------------------------------------------------------------
// Host launcher
// ---------------------------------------------------------------------------
extern "C" void kernel_launch(void* const* d_in, const int* in_sizes, int n_in,
                              void* d_out, int out_size, void* d_ws,
                              size_t ws_size, hipStream_t stream) {
  (void)in_sizes; (void)n_in; (void)out_size; (void)ws_size;
  const float* sites    = (const float*)d_in[0];
  const float* bonds    = (const float*)d_in[1];
  const float* sites_p  = (const float*)d_in[2];
  const float* bonds_sp = (const float*)d_in[3];
  const float* bonds_ps = (const float*)d_in[4];
  const int* idx1    = (const int*)d_in[5];
  const int* idx2    = (const int*)d_in[6];
  const int* idx1_sp = (const int*)d_in[7];
  const int* idx2_sp = (const int*)d_in[8];
  const int* idx1_ps = (const int*)d_in[9];
  const int* idx2_ps = (const int*)d_in[10];
  const float* se_W  = (const float*)d_in[11];
  const float* se_b  = (const float*)d_in[12];
  const float* sep_W = (const float*)d_in[13];
  const float* sep_b = (const float*)d_in[14];
  const float* ee_W  = (const float*)d_in[15];
  const float* ee_b  = (const float*)d_in[16];
  const float* eep_W = (const float*)d_in[17];
  const float* eep_b = (const float*)d_in[18];
  // type order: 0=ss (19..23), 1=ps (24..28), 2=sp (29..33)
  const float* eqW1[3] = {(const float*)d_in[19], (const float*)d_in[24], (const float*)d_in[29]};
  const float* eqW2[3] = {(const float*)d_in[20], (const float*)d_in[25], (const float*)d_in[30]};
  const float* eqB[3]  = {(const float*)d_in[21], (const float*)d_in[26], (const float*)d_in[31]};
  const float* atW[3]  = {(const float*)d_in[22], (const float*)d_in[27], (const float*)d_in[32]};
  const float* atB[3]  = {(const float*)d_in[23], (const float*)d_in[28], (const float*)d_in[33]};
  const float* nu_W1  = (const float*)d_in[34];
  const float* nu_b1  = (const float*)d_in[35];
  const float* nu_W2  = (const float*)d_in[36];
  const float* nu_b2  = (const float*)d_in[37];
  const float* nup_W1 = (const float*)d_in[38];
  const float* nup_b1 = (const float*)d_in[39];
  const float* nup_W2 = (const float*)d_in[40];
  const float* nup_b2 = (const float*)d_in[41];
  const float* p1_W  = (const float*)d_in[42];
  const float* p1_b  = (const float*)d_in[43];
  const float* p2a_W = (const float*)d_in[44];
  const float* p2a_b = (const float*)d_in[45];
  const float* p2b_W = (const float*)d_in[46];
  const float* p2b_b = (const float*)d_in[47];
  const float* p3_W  = (const float*)d_in[48];
  const float* p3_b  = (const float*)d_in[49];

  // workspace carve (~1.04 MB of f32)
  float* ws = (float*)d_ws;
  float* s_ws   = ws; ws += 4 * 128 * 32;   // site features
  float* sp_ws  = ws; ws += 4 * 16 * 32;    // pore features
  float* be_ws  = ws; ws += 4 * 1280 * 16;  // ss edge embeddings
  float* bsp_ws = ws; ws += 4 * 1024 * 16;  // sp edge embeddings
  float* bps_ws = ws; ws += 4 * 1024 * 16;  // ps edge embeddings
  float* m_ws   = ws; ws += 4 * 128 * 32;   // ss messages -> sites
  float* mps_ws = ws; ws += 4 * 128 * 32;   // ps messages -> sites
  float* msp_ws = ws; ws += 4 * 16 * 32;    // sp messages -> pores

  MPNN_embed_sites<<<72, 256, 0, stream>>>(sites, sites_p, se_W, se_b, sep_W,
                                           sep_b, s_ws, sp_ws);
  MPNN_embed_edges<<<104, 256, 0, stream>>>(bonds, bonds_sp, bonds_ps, ee_W,
                                            ee_b, eep_W, eep_b, be_ws, bsp_ws,
                                            bps_ws);
  for (int t = 0; t < 3; ++t) {
    // ss: v = [s[idx1] | s[idx2] | be], recv = sites
    MPNN_msg<<<4, 256, 0, stream>>>(s_ws, s_ws, be_ws, idx1, idx2,
        eqW1[0] + t * 2560, eqW2[0] + t * 2560, eqB[0] + t * 32,
        atW[0] + t * 32, atB[0] + t, m_ws, 1280, 128, 128, 128);
    // ps: v = [sp[idx1_ps] | s[idx2_ps] | bps], recv = sites
    MPNN_msg<<<4, 256, 0, stream>>>(sp_ws, s_ws, bps_ws, idx1_ps, idx2_ps,
        eqW1[1] + t * 2560, eqW2[1] + t * 2560, eqB[1] + t * 32,
        atW[1] + t * 32, atB[1] + t, mps_ws, 1024, 16, 128, 128);
    // sp: v = [s[idx1_sp] | sp[idx2_sp] | bsp], recv = pores
    MPNN_msg<<<4, 256, 0, stream>>>(s_ws, sp_ws, bsp_ws, idx1_sp, idx2_sp,
        eqW1[2] + t * 2560, eqW2[2] + t * 2560, eqB[2] + t * 32,
        atW[2] + t * 32, atB[2] + t, msp_ws, 1024, 128, 16, 16);
    MPNN_update<<<8, 256, 0, stream>>>(s_ws, sp_ws, m_ws, mps_ws, msp_ws,
        nu_W1 + t * 3072, nu_b1 + t * 32, nu_W2 + t * 1024, nu_b2 + t * 32,
        nup_W1 + t * 2048, nup_b1 + t * 32, nup_W2 + t * 1024, nup_b2 + t * 32);
  }
  MPNN_head<<<1, 256, 0, stream>>>(s_ws, p1_W, p1_b, p2a_W, p2a_b, p2b_W,
                                   p2b_b, p3_W, p3_b, (float*)d_out);
}